// RipsModel_9861244911651
// MI455X (gfx1250) — compile-verified
//
#include <hip/hip_runtime.h>

// CDNA5 / gfx1250: batched 64-dim distance evaluation for 100 index pairs
// using V_WMMA_F32_16X16X4_F32 (full fp32 precision, matches reference math).

typedef __attribute__((ext_vector_type(2))) float v2f;
typedef __attribute__((ext_vector_type(8))) float v8f;

#define CARD   50
#define NPAIRS (2 * CARD)   // 100 distances to produce
#define DIMS   64           // feature dimension of X

__global__ __launch_bounds__(32) void rips_pairs_wmma(
    const float* __restrict__ X,   // [8192, 64] fp32
    const int*   __restrict__ ids, // [2*NPAIRS] int32, pair p = (ids[2p], ids[2p+1])
    float*       __restrict__ out) // [NPAIRS] fp32
{
  const int lane = threadIdx.x;        // 0..31, one full wave per block
  const int m    = lane & 15;          // row/col index inside the 16x16 tile
  const int kh   = (lane >> 4) << 1;   // K sub-offset: 0 (lanes 0-15) / 2 (lanes 16-31)

  // This lane owns pair p's rows for fragment slot m (clamped for the tail tile;
  // clamp is data-only, no divergence, EXEC stays all-1s for WMMA).
  int p  = blockIdx.x * 16 + m;
  int pc = (p < NPAIRS) ? p : (NPAIRS - 1);
  const int ia = ids[2 * pc + 0];
  const int ib = ids[2 * pc + 1];
  const float* Xa = X + (long)ia * DIMS;
  const float* Xb = X + (long)ib * DIMS;

  // Three accumulation chains: diag(cab)=<a,b>, diag(caa)=||a||^2, diag(cbb)=||b||^2.
  v8f cab = {}, caa = {}, cbb = {};
#pragma unroll
  for (int c = 0; c < DIMS / 4; ++c) {
    const int k = c * 4 + kh;
    // Same lane->(index,K) mapping for A and B fragments of 32-bit WMMA,
    // so one loaded pair serves as both "A row" and "B column".
    v2f a = *(const v2f*)(Xa + k);   // 8B-aligned: rows are 256B, k even
    v2f b = *(const v2f*)(Xb + k);
    cab = __builtin_amdgcn_wmma_f32_16x16x4_f32(false, a, false, b, (short)0, cab, false, false);
    caa = __builtin_amdgcn_wmma_f32_16x16x4_f32(false, a, false, a, (short)0, caa, false, false);
    cbb = __builtin_amdgcn_wmma_f32_16x16x4_f32(false, b, false, b, (short)0, cbb, false, false);
  }

  // Diagonal extraction per the 16x16 f32 C/D layout:
  //   D[p][p], p<8 : lane p,     VGPR p
  //   D[p][p], p>=8: lane p+16,  VGPR p-8
  const bool hi    = (lane >= 24);
  const bool valid = (lane < 8) || hi;
  const int  idx   = lane & 7;
  const float sab = cab[idx];
  const float saa = caa[idx];
  const float sbb = cbb[idx];

  if (valid) {
    const int pd = hi ? (lane - 16) : lane;   // local diagonal index 0..15
    const int pg = blockIdx.x * 16 + pd;      // global pair index
    if (pg < NPAIRS) {
      float d2 = saa + sbb - 2.0f * sab;
      d2 = fmaxf(d2, 0.0f);
      // reference: zero entries (incl. i==j) forced to exactly 0
      out[pg] = (d2 > 0.0f) ? __builtin_sqrtf(d2) : 0.0f;
    }
  }
}

extern "C" void kernel_launch(void* const* d_in, const int* in_sizes, int n_in,
                              void* d_out, int out_size, void* d_ws, size_t ws_size,
                              hipStream_t stream) {
  const float* X   = (const float*)d_in[0];  // [8192*64] fp32
  const int*   ids = (const int*)d_in[1];    // [200] int32
  float*       out = (float*)d_out;          // [100] fp32

  // 7 waves cover ceil(100/16) tiles of 16 pairs each; one wave per block.
  dim3 grid((NPAIRS + 15) / 16);
  dim3 block(32);
  rips_pairs_wmma<<<grid, block, 0, stream>>>(X, ids, out);
}